// DiceLoss_80470507258049
// MI455X (gfx1250) — compile-verified
//
#include <hip/hip_runtime.h>
#include <hip/hip_bf16.h>

#define NUM_ORGAN 13
#define NCH 14
#define VOX (48 * 256 * 256)          // 3,145,728 voxels per (b, channel)
#define V4  (VOX / 4)                 // 786,432 float4 elements
#define EPS 1e-5

typedef __attribute__((ext_vector_type(2))) float v2f;
typedef __attribute__((ext_vector_type(8))) float v8f;

// Full 32-lane sum using the CDNA5 matrix pipe.
// V_WMMA_F32_16X16X4_F32:  A(16x4) layout: lanes 0-15 hold {K0,K1} in vgpr{0,1},
// lanes 16-31 hold {K2,K3}.  With A = {x, 0} and B = all-ones:
//   D[m][n] = x_m + x_{m+16}   (independent of n)
// D vgpr r: lanes 0-15 carry M=r, lanes 16-31 carry M=8+r.  Summing the 8 D
// vgprs per lane gives the half-wave totals; one xor-16 shuffle completes it.
__device__ __forceinline__ float wave_sum_f32(float x) {
    v2f a; a.x = x;    a.y = 0.0f;
    v2f b; b.x = 1.0f; b.y = 1.0f;
    v8f c = {};
    v8f d = __builtin_amdgcn_wmma_f32_16x16x4_f32(
        /*neg_a=*/false, a, /*neg_b=*/false, b,
        /*c_mod=*/(short)0, c, /*reuse_a=*/false, /*reuse_b=*/false);
    float t = d[0] + d[1] + d[2] + d[3] + d[4] + d[5] + d[6] + d[7];
    t += __shfl_xor(t, 16, 32);   // combine the two half-wave totals
    return t;                     // every lane now holds the full 32-lane sum
}

// ws layout: double[b][organ][5] = {inter1, sq1, inter2, sq2, cnt}
__global__ void dice_partial_kernel(const float* __restrict__ p1,
                                    const float* __restrict__ p2,
                                    const int*   __restrict__ tgt,
                                    double*      __restrict__ ws)
{
    const int b   = blockIdx.y;
    const int g   = blockIdx.x * blockDim.x + threadIdx.x;
    const int nth = gridDim.x * blockDim.x;

    const int4*   t4 = reinterpret_cast<const int4*>(tgt) + (size_t)b * V4;
    const float4* a4 = reinterpret_cast<const float4*>(p1) + ((size_t)b * NCH + 1) * (size_t)V4;
    const float4* b4 = reinterpret_cast<const float4*>(p2) + ((size_t)b * NCH + 1) * (size_t)V4;

    float inter1[NUM_ORGAN], sq1[NUM_ORGAN], inter2[NUM_ORGAN], sq2[NUM_ORGAN], cnt[NUM_ORGAN];
#pragma unroll
    for (int o = 0; o < NUM_ORGAN; ++o) {
        inter1[o] = 0.f; sq1[o] = 0.f; inter2[o] = 0.f; sq2[o] = 0.f; cnt[o] = 0.f;
    }

    // 512 blocks * 256 threads per batch -> exactly 6 iterations per thread;
    // no divergence, EXEC stays all-1s for the WMMA epilogue.
    for (int i = g; i < V4; i += nth) {
        const int4 t = t4[i];
#pragma unroll
        for (int o = 0; o < NUM_ORGAN; ++o) {
            const int organ = o + 1;
            const float4 p = a4[(size_t)o * V4 + i];
            const float4 q = b4[(size_t)o * V4 + i];
            const float m0 = (t.x == organ) ? 1.f : 0.f;
            const float m1 = (t.y == organ) ? 1.f : 0.f;
            const float m2 = (t.z == organ) ? 1.f : 0.f;
            const float m3 = (t.w == organ) ? 1.f : 0.f;
            inter1[o] += p.x * m0 + p.y * m1 + p.z * m2 + p.w * m3;
            sq1[o]    += p.x * p.x + p.y * p.y + p.z * p.z + p.w * p.w;
            inter2[o] += q.x * m0 + q.y * m1 + q.z * m2 + q.w * m3;
            sq2[o]    += q.x * q.x + q.y * q.y + q.z * q.z + q.w * q.w;
            cnt[o]    += m0 + m1 + m2 + m3;
        }
    }

    const int lane = threadIdx.x & 31;
#pragma unroll
    for (int o = 0; o < NUM_ORGAN; ++o) {
        const float s0 = wave_sum_f32(inter1[o]);
        const float s1 = wave_sum_f32(sq1[o]);
        const float s2 = wave_sum_f32(inter2[o]);
        const float s3 = wave_sum_f32(sq2[o]);
        const float s4 = wave_sum_f32(cnt[o]);
        if (lane == 0) {
            double* slot = ws + ((size_t)b * NUM_ORGAN + o) * 5;
            atomicAdd(slot + 0, (double)s0);
            atomicAdd(slot + 1, (double)s1);
            atomicAdd(slot + 2, (double)s2);
            atomicAdd(slot + 3, (double)s3);
            atomicAdd(slot + 4, (double)s4);
        }
    }
}

__global__ void dice_finalize_kernel(const double* __restrict__ ws,
                                     float* __restrict__ out)
{
    if (threadIdx.x == 0 && blockIdx.x == 0) {
        double loss = 0.0;
        for (int b = 0; b < 2; ++b) {
            double dsum = 0.0;
            for (int o = 0; o < NUM_ORGAN; ++o) {
                const double* s = ws + ((size_t)b * NUM_ORGAN + o) * 5;
                const double inter1 = s[0], sq1 = s[1];
                const double inter2 = s[2], sq2 = s[3], t2 = s[4];
                dsum += 2.0 * inter1 / (sq1 + t2 + EPS);
                dsum += 2.0 * inter2 / (sq2 + t2 + EPS);
            }
            loss += 2.0 - dsum / (double)NUM_ORGAN;
        }
        out[0] = (float)(loss * 0.5);   // mean over batch of 2
    }
}

extern "C" void kernel_launch(void* const* d_in, const int* in_sizes, int n_in,
                              void* d_out, int out_size, void* d_ws, size_t ws_size,
                              hipStream_t stream) {
    const float* pred1 = (const float*)d_in[0];
    const float* pred2 = (const float*)d_in[1];
    const int*   tgt   = (const int*)d_in[2];
    float*  out = (float*)d_out;
    double* acc = (double*)d_ws;

    (void)in_sizes; (void)n_in; (void)out_size; (void)ws_size;

    // Zero the 2*13*5 double accumulators (async memset is graph-capturable).
    hipMemsetAsync(acc, 0, (size_t)2 * NUM_ORGAN * 5 * sizeof(double), stream);

    dim3 grid(512, 2);
    dice_partial_kernel<<<grid, 256, 0, stream>>>(pred1, pred2, tgt, acc);
    dice_finalize_kernel<<<1, 32, 0, stream>>>(acc, out);
}